// ContactMapGenerator_79078937854413
// MI455X (gfx1250) — compile-verified
//
#include <hip/hip_runtime.h>
#include <stdint.h>

// Problem constants (from setup_inputs in the reference)
#define B    16
#define N    131072
#define NP   1024
#define FPT  512            // threads per FPS workgroup (16 wave32 waves)
#define PPT  (N / FPT)      // 256 points owned per thread
#define RREG 128            // distances kept in VGPRs per thread (j = 0..127)
#define LGRP 18             // float4 LDS groups per thread (72 dists, j = 128..199)
#define WGRP 14             // float4 ws  groups per thread (56 dists, j = 200..255)
// RREG + 4*LGRP + 4*WGRP == PPT

struct F3 { float x, y, z; };

// Match XLA's square-then-sum without FMA contraction surprises.
__device__ __forceinline__ float sq3(float dx, float dy, float dz) {
  return __fadd_rn(__fadd_rn(__fmul_rn(dx, dx), __fmul_rn(dy, dy)),
                   __fmul_rn(dz, dz));
}

typedef unsigned int u32;
typedef u32 v4u __attribute__((ext_vector_type(4)));
typedef int v4i __attribute__((ext_vector_type(4)));
typedef int v8i __attribute__((ext_vector_type(8)));

// ---------------------------------------------------------------------------
// FPS: one 512-thread workgroup per batch (4 waves/SIMD -> 256 VGPR budget).
// Running-min distances split 128-VGPR / 72-LDS / 56-L2(ws) per thread.
// Mesh (1.5 MB/batch) re-streams from the 192 MB L2 each iteration.
// ---------------------------------------------------------------------------
__global__ __launch_bounds__(FPT) void fps_kernel(
    const float* __restrict__ mesh, const int* __restrict__ finit,
    float* __restrict__ idx_out_f, int* __restrict__ idx_ws,
    float4* __restrict__ wdist) {
  const int b = blockIdx.x, t = threadIdx.x;
  const float* m = mesh + (size_t)b * N * 3;
  float4* wd = wdist + (size_t)b * (WGRP * FPT);

  __shared__ float4 s_dist[LGRP * FPT];  // 144 KB, thread-private slots
  __shared__ float  s_val[16];
  __shared__ int    s_idx[16];
  __shared__ int    s_cur;

#if defined(__has_builtin)
#if __has_builtin(__builtin_amdgcn_tensor_load_to_lds)
  {
    // CDNA5 Tensor Data Mover: async-load the first 32 KB of this batch's
    // mesh into the (not-yet-initialized) LDS distance buffer — pure
    // L2/LDS warm-up, overwritten by the init below. D# per ISA ch.8.
    uint64_t ga  = (uint64_t)(uintptr_t)m;
    u32      lds = (u32)(uintptr_t)(&s_dist[0]);
    v4u g0 = { 1u, lds, (u32)ga,
               (u32)((ga >> 32) & 0x1FFFFFFu) | 0x80000000u };
    v8i g1 = { (int)(2u << 16),                    // data_size = 4 bytes
               (int)((8192u & 0xFFFFu) << 16),     // tensor_dim0 lo16
               (int)((8192u >> 16) | (1u << 16)),  // dim0 hi16 | tensor_dim1=1
               (int)(8192u << 16),                 // tile_dim0 = 8192
               0,                                  // tile_dim1/2 unused
               (int)8192u,                         // tensor_dim0_stride lo32
               0, 0 };
    v4i g2 = {0, 0, 0, 0};
    v4i g3 = {0, 0, 0, 0};
#if defined(__clang_major__) && __clang_major__ >= 23
    v8i g4 = {0, 0, 0, 0, 0, 0, 0, 0};
    __builtin_amdgcn_tensor_load_to_lds(g0, g1, g2, g3, g4, 0);
#else
    __builtin_amdgcn_tensor_load_to_lds(g0, g1, g2, g3, 0);
#endif
    __builtin_amdgcn_s_wait_tensorcnt(0);
  }
#endif
#endif
  __syncthreads();  // all waves' TDM writes done before dist init

  const float4 finit4 = make_float4(1e10f, 1e10f, 1e10f, 1e10f);
  float dist[RREG];
#pragma unroll
  for (int j = 0; j < RREG; ++j) dist[j] = 1e10f;
#pragma unroll
  for (int g = 0; g < LGRP; ++g) s_dist[g * FPT + t] = finit4;
  for (int g = 0; g < WGRP; ++g) wd[g * FPT + t] = finit4;

  const float* mt = m + 3 * t;  // thread-coalesced b96 loads, imm offsets
  const int lane = t & 31, wv = t >> 5;
  int cur = __builtin_amdgcn_readfirstlane(finit[b]);

  for (int it = 0; it < NP; ++it) {
    // Reference's lax.scan emits the *incoming* farthest each step.
    if (t == 0) {
      idx_out_f[b * NP + it] = (float)cur;
      idx_ws[b * NP + it]    = cur;
    }
    if (it == NP - 1) break;  // final argmax is discarded by the reference

    // Uniform centroid fetch (scalar path via readfirstlane'd index).
    const float cx = m[3 * cur + 0];
    const float cy = m[3 * cur + 1];
    const float cz = m[3 * cur + 2];

    float bv = -1.0f; int bi = 0;

    // ---- VGPR-resident: global indices t + j*512, j = 0..127 ----
#pragma unroll
    for (int j = 0; j < RREG; ++j) {
      F3 p = *(const F3*)(mt + 3 * FPT * j);
      float d  = sq3(p.x - cx, p.y - cy, p.z - cz);
      float nd = fminf(dist[j], d);
      dist[j] = nd;
      if (nd > bv) { bv = nd; bi = t + (j << 9); }  // strict > => lowest idx
    }

    // ---- LDS-resident: j = 128..199, vectorized ds_b128 ----
#pragma unroll
    for (int g = 0; g < LGRP; ++g) {
      float4 dv = s_dist[g * FPT + t];
      float nd[4];
#pragma unroll
      for (int k = 0; k < 4; ++k) {
        const int j = RREG + 4 * g + k;
        F3 p = *(const F3*)(mt + 3 * FPT * j);
        float dk = (k == 0) ? dv.x : (k == 1) ? dv.y : (k == 2) ? dv.z : dv.w;
        nd[k] = fminf(dk, sq3(p.x - cx, p.y - cy, p.z - cz));
        if (nd[k] > bv) { bv = nd[k]; bi = t + (j << 9); }
      }
      s_dist[g * FPT + t] = make_float4(nd[0], nd[1], nd[2], nd[3]);
    }

    // ---- L2/ws-resident: j = 200..255, coalesced global b128 ----
#pragma unroll
    for (int g = 0; g < WGRP; ++g) {
      float4 dv = wd[g * FPT + t];
      float nd[4];
#pragma unroll
      for (int k = 0; k < 4; ++k) {
        const int j = RREG + 4 * LGRP + 4 * g + k;
        F3 p = *(const F3*)(mt + 3 * FPT * j);
        float dk = (k == 0) ? dv.x : (k == 1) ? dv.y : (k == 2) ? dv.z : dv.w;
        nd[k] = fminf(dk, sq3(p.x - cx, p.y - cy, p.z - cz));
        if (nd[k] > bv) { bv = nd[k]; bi = t + (j << 9); }
      }
      wd[g * FPT + t] = make_float4(nd[0], nd[1], nd[2], nd[3]);
    }

    // wave32 argmax reduce, first-index tie-break (matches jnp.argmax)
#pragma unroll
    for (int o = 16; o; o >>= 1) {
      float ov = __shfl_down(bv, o, 32);
      int   oi = __shfl_down(bi, o, 32);
      if (ov > bv || (ov == bv && oi < bi)) { bv = ov; bi = oi; }
    }
    if (lane == 0) { s_val[wv] = bv; s_idx[wv] = bi; }
    __syncthreads();
    if (wv == 0) {
      if (lane < 16) { bv = s_val[lane]; bi = s_idx[lane]; }
      else           { bv = -1.0f;       bi = 0x7fffffff; }
#pragma unroll
      for (int o = 16; o; o >>= 1) {
        float ov = __shfl_down(bv, o, 32);
        int   oi = __shfl_down(bi, o, 32);
        if (ov > bv || (ov == bv && oi < bi)) { bv = ov; bi = oi; }
      }
      if (lane == 0) s_cur = bi;
    }
    __syncthreads();
    cur = __builtin_amdgcn_readfirstlane(s_cur);
  }
}

// ---------------------------------------------------------------------------
// Per-batch centroid + object_scale = max ||p - centroid||
// ---------------------------------------------------------------------------
__global__ __launch_bounds__(256) void scale_kernel(
    const float* __restrict__ mesh, float* __restrict__ scale_out) {
  const int b = blockIdx.x, t = threadIdx.x;
  const float* m = mesh + (size_t)b * N * 3;
  __shared__ float sx[256], sy[256], sz[256];
  __shared__ float c[3];

  float ax = 0.f, ay = 0.f, az = 0.f;
  for (int i = t; i < N; i += 256) {
    F3 p = ((const F3*)m)[i];
    ax += p.x; ay += p.y; az += p.z;
  }
  sx[t] = ax; sy[t] = ay; sz[t] = az;
  __syncthreads();
  for (int o = 128; o; o >>= 1) {
    if (t < o) { sx[t] += sx[t + o]; sy[t] += sy[t + o]; sz[t] += sz[t + o]; }
    __syncthreads();
  }
  if (t == 0) { c[0] = sx[0] / N; c[1] = sy[0] / N; c[2] = sz[0] / N; }
  __syncthreads();

  const float cx = c[0], cy = c[1], cz = c[2];
  float mx = 0.f;
  for (int i = t; i < N; i += 256) {
    F3 p = ((const F3*)m)[i];
    mx = fmaxf(mx, sq3(p.x - cx, p.y - cy, p.z - cz));
  }
  sx[t] = mx;
  __syncthreads();
  for (int o = 128; o; o >>= 1) {
    if (t < o) sx[t] = fmaxf(sx[t], sx[t + o]);
    __syncthreads();
  }
  // max(sqrt(d)) == sqrt(max(d)) (monotone), so one sqrt suffices.
  if (t == 0) scale_out[b] = sqrtf(sx[0]);
}

// ---------------------------------------------------------------------------
// Gather sampled points, normalize by object scale
// ---------------------------------------------------------------------------
__global__ __launch_bounds__(256) void gather_kernel(
    const float* __restrict__ mesh, const int* __restrict__ idx_ws,
    const float* __restrict__ scales, float* __restrict__ pc) {
  const int i = blockIdx.x * 256 + threadIdx.x;  // 0 .. B*NP-1
  const int b = i >> 10;
  const int id = idx_ws[i];
  const float s = scales[b];
  const float* p = mesh + ((size_t)b * N + id) * 3;
  pc[3 * i + 0] = p[0] / s;
  pc[3 * i + 1] = p[1] / s;
  pc[3 * i + 2] = p[2] / s;
}

// ---------------------------------------------------------------------------
extern "C" void kernel_launch(void* const* d_in, const int* in_sizes, int n_in,
                              void* d_out, int out_size, void* d_ws,
                              size_t ws_size, hipStream_t stream) {
  const float* mesh  = (const float*)d_in[0];
  const int*   finit = (const int*)d_in[1];
  // npoints (d_in[2]) is the compile-time constant NP = 1024.

  float* pc     = (float*)d_out;            // [B, NP, 3]
  float* scales = pc + (size_t)B * NP * 3;  // [B]
  float* idxf   = scales + B;               // [B, NP] (as float values)

  int*    idx_ws = (int*)d_ws;              // [B, NP] integer scratch (64 KB)
  float4* wdist  = (float4*)((char*)d_ws + (size_t)B * NP * sizeof(int));
  // wdist: [B][WGRP*FPT] float4 = 1.75 MB running-min spill tier (L2-resident)

  scale_kernel<<<B, 256, 0, stream>>>(mesh, scales);
  fps_kernel<<<B, FPT, 0, stream>>>(mesh, finit, idxf, idx_ws, wdist);
  gather_kernel<<<(B * NP) / 256, 256, 0, stream>>>(mesh, idx_ws, scales, pc);
}